// LFMMILoss_47038481826030
// MI455X (gfx1250) — compile-verified
//
#include <hip/hip_runtime.h>
#include <hip/hip_bf16.h>

typedef __attribute__((ext_vector_type(2))) float v2f;
typedef __attribute__((ext_vector_type(8))) float v8f;

#define BB 16
#define TT 500
#define CC 2048
#define SN 64
#define SD 512

__device__ __forceinline__ float emclip(float v) {
    return fminf(fmaxf(v, -30.0f), 30.0f);
}

// ---- Kernel 1: pair-interleaved exp(den_logA):  (k,n) -> (k>>1)*1024 + 2n + (k&1)
// Lets the den kernel fetch both K-values of a WMMA B-fragment with one b64 load.
__global__ void prep_expA(const float* __restrict__ logA, float* __restrict__ expAP) {
    int idx = blockIdx.x * blockDim.x + threadIdx.x;
    if (idx < SD * SD) {
        int i = idx >> 9;    // row k
        int j = idx & 511;   // col n
        expAP[(size_t)((i >> 1) * SD + j) * 2 + (i & 1)] = __expf(logA[idx]);
    }
}

// ---- Kernel 2: numerator forward, one block per batch, 64 threads (state j) ----
__global__ __launch_bounds__(SN) void num_fwd(
    const float* __restrict__ x, const int* __restrict__ seqlen,
    const float* __restrict__ logA, const float* __restrict__ lstart,
    const float* __restrict__ lfinal, const int* __restrict__ pdf,
    float* __restrict__ partial)
{
    __shared__ float alpha[SN];
    __shared__ float v[SN];
    const int b = blockIdx.x, j = threadIdx.x;
    const float* xb = x + (size_t)b * TT * CC;
    const float* Ab = logA + (size_t)b * SN * SN;
    const int pj = pdf[b * SN + j];
    alpha[j] = lstart[b * SN + j] + emclip(xb[pj]);
    __syncthreads();
    const int L = seqlen[b];
    for (int t = 1; t < TT; ++t) {
        float mx = -3.4e38f;
        #pragma unroll 8
        for (int i = 0; i < SN; ++i) mx = fmaxf(mx, alpha[i] + Ab[i * SN + j]);
        float s = 0.0f;
        #pragma unroll 8
        for (int i = 0; i < SN; ++i) s += __expf(alpha[i] + Ab[i * SN + j] - mx);
        float nv = mx + __logf(s) + emclip(xb[(size_t)t * CC + pj]);
        __syncthreads();
        if (t < L) alpha[j] = nv;
        __syncthreads();
    }
    v[j] = alpha[j] + lfinal[b * SN + j];
    __syncthreads();
    if (j == 0) {
        float mx = -3.4e38f;
        for (int i = 0; i < SN; ++i) mx = fmaxf(mx, v[i]);
        float s = 0.0f;
        for (int i = 0; i < SN; ++i) s += __expf(v[i] - mx);
        partial[1 + b] = mx + __logf(s);
    }
}

// ---- Kernel 3: denominator forward, state kept as (p, m):
//      p[16][512] = exp(alpha - m) in LDS, m[16] per-row log offsets.
//      Per step: q = p * expA (WMMA f32 16x16x4), val = log(q)+em written in-place,
//      then per-row renormalize: m += max(val), p = exp(val - max). ----
__global__ __launch_bounds__(1024) void den_fwd(
    const float* __restrict__ x, const int* __restrict__ seqlen,
    const float* __restrict__ expAP, const float* __restrict__ lstart,
    const float* __restrict__ lfinal, const int* __restrict__ pdf,
    float* __restrict__ partial)
{
    __shared__ float pbuf[BB][SD];   // 32 KB state (p); holds log-vals transiently
    __shared__ float wred[32];
    __shared__ float m[BB];
    __shared__ float dl[BB];
    __shared__ float llh[BB];
    __shared__ int   sl[BB];

    const int tid  = threadIdx.x;
    const int b    = tid >> 6;        // row owned for reductions (uniform per wave)
    const int k    = tid & 63;        // 8 columns each
    const int lane = tid & 31;
    const int wave = tid >> 5;        // 0..31 -> N tile
    const int N0   = wave * 16;
    const int Ml   = lane & 15;       // M (A-frag) / N offset (D-frag)
    const int hi   = lane >> 4;       // K-pair select / M-half select

    if (tid < BB) { sl[tid] = seqlen[tid]; m[tid] = 0.0f; }
    // t=0: log-domain init values into pbuf
    #pragma unroll
    for (int u = 0; u < 8; ++u) {
        int j = k * 8 + u;
        pbuf[b][j] = lstart[j] + emclip(x[(size_t)b * TT * CC + pdf[j]]);
    }
    __syncthreads();

    // initial normalization: m = rowmax, p = exp(val - rowmax)
    {
        float mx = -3.4e38f;
        #pragma unroll
        for (int u = 0; u < 8; ++u) mx = fmaxf(mx, pbuf[b][k * 8 + u]);
        #pragma unroll
        for (int off = 16; off > 0; off >>= 1) mx = fmaxf(mx, __shfl_xor(mx, off, 32));
        if (lane == 0) wred[wave] = mx;
        __syncthreads();
        if (tid < BB) {
            float d = fmaxf(wred[2 * tid], wred[2 * tid + 1]);
            m[tid] += d; dl[tid] = d;
        }
        __syncthreads();
        float d = dl[b];
        #pragma unroll
        for (int u = 0; u < 8; ++u) { int j = k * 8 + u; pbuf[b][j] = __expf(pbuf[b][j] - d); }
        __syncthreads();
    }

    for (int t = 1; t < TT; ++t) {
        // --- WMMA phase: D[0:16, N0:N0+16] = p (16x512) * expA (512x512) ---
        v8f c = {0.f, 0.f, 0.f, 0.f, 0.f, 0.f, 0.f, 0.f};
        const float* Arow = &pbuf[Ml][hi * 2];                       // + kk*4
        const float* Bp   = expAP + hi * (SD * 2) + (size_t)(N0 + Ml) * 2;  // + kk*2048
        #pragma unroll 4
        for (int kk = 0; kk < SD / 4; ++kk) {
            v2f a  = *(const v2f*)(Arow + kk * 4);        // ds_load_b64
            v2f bf = *(const v2f*)(Bp + (size_t)kk * (SD * 4)); // global_load_b64
            c = __builtin_amdgcn_wmma_f32_16x16x4_f32(
                    false, a, false, bf, (short)0, c, false, false);
        }
        __syncthreads();   // all p reads complete before in-place overwrite

        // --- writeback (in place, unmasked rows only): val = log(q) + em ---
        #pragma unroll
        for (int r = 0; r < 8; ++r) {
            const int M = r + hi * 8;
            const int N = N0 + Ml;
            if (t < sl[M]) {
                float em = emclip(x[(size_t)M * TT * CC + (size_t)t * CC + pdf[N]]);
                pbuf[M][N] = __logf(fmaxf(c[r], 1e-37f)) + em;
            }
        }
        __syncthreads();

        // --- renormalize unmasked rows: m += rowmax(val), p = exp(val - rowmax) ---
        const bool act = (t < sl[b]);   // uniform per wave
        float mx = -3.4e38f;
        if (act) {
            #pragma unroll
            for (int u = 0; u < 8; ++u) mx = fmaxf(mx, pbuf[b][k * 8 + u]);
        }
        #pragma unroll
        for (int off = 16; off > 0; off >>= 1) mx = fmaxf(mx, __shfl_xor(mx, off, 32));
        if (lane == 0) wred[wave] = mx;
        __syncthreads();
        if (tid < BB) {
            float d = fmaxf(wred[2 * tid], wred[2 * tid + 1]);
            if (t < sl[tid]) m[tid] += d;
            dl[tid] = d;
        }
        __syncthreads();
        if (act) {
            float d = dl[b];
            #pragma unroll
            for (int u = 0; u < 8; ++u) { int j = k * 8 + u; pbuf[b][j] = __expf(pbuf[b][j] - d); }
        }
        __syncthreads();
    }

    // --- final: llh_b = m_b + log( sum_j p[b][j] * exp(lfinal[j]) ) ---
    float s = 0.0f;
    #pragma unroll
    for (int u = 0; u < 8; ++u) {
        int j = k * 8 + u;
        s += pbuf[b][j] * __expf(lfinal[j]);
    }
    #pragma unroll
    for (int off = 16; off > 0; off >>= 1) s += __shfl_xor(s, off, 32);
    if (lane == 0) wred[wave] = s;
    __syncthreads();
    if (tid < BB) {
        float ss = wred[2 * tid] + wred[2 * tid + 1];
        llh[tid] = m[tid] + __logf(fmaxf(ss, 1e-37f));
    }
    __syncthreads();
    if (tid == 0) {
        float d = 0.0f;
        for (int q = 0; q < BB; ++q) d += llh[q];
        partial[0] = d;
    }
}

// ---- Kernel 4: loss = den_llh - num_llh (DEN_SCALE = 1) ----
__global__ void finish_k(const float* __restrict__ partial, float* __restrict__ out) {
    float num = 0.0f;
    for (int b = 0; b < BB; ++b) num += partial[1 + b];
    out[0] = partial[0] - num;
}

extern "C" void kernel_launch(void* const* d_in, const int* in_sizes, int n_in,
                              void* d_out, int out_size, void* d_ws, size_t ws_size,
                              hipStream_t stream) {
    const float* x        = (const float*)d_in[0];
    const int*   seqlen   = (const int*)  d_in[1];
    const float* num_logA = (const float*)d_in[2];
    const float* num_ls   = (const float*)d_in[3];
    const float* num_lf   = (const float*)d_in[4];
    const int*   num_pdf  = (const int*)  d_in[5];
    const float* den_logA = (const float*)d_in[6];
    const float* den_ls   = (const float*)d_in[7];
    const float* den_lf   = (const float*)d_in[8];
    const int*   den_pdf  = (const int*)  d_in[9];

    float* ws      = (float*)d_ws;
    float* expAP   = ws;               // 512*512 floats = 1 MB, pair-interleaved
    float* partial = ws + SD * SD;     // [0]=den_llh, [1..16]=num_llh_b

    prep_expA<<<(SD * SD + 255) / 256, 256, 0, stream>>>(den_logA, expAP);
    num_fwd<<<BB, SN, 0, stream>>>(x, seqlen, num_logA, num_ls, num_lf, num_pdf, partial);
    den_fwd<<<1, 1024, 0, stream>>>(x, seqlen, expAP, den_ls, den_lf, den_pdf, partial);
    finish_k<<<1, 1, 0, stream>>>(partial, (float*)d_out);
}